// HashGridEncoder_36378372997412
// MI455X (gfx1250) — compile-verified
//
#include <hip/hip_runtime.h>
#include <cmath>
#include <cstdint>

// ---------------- compile-time constants matching the reference ----------------
#define BLOCK   256          // 8 waves (wave32)
#define ITER    16           // points per thread -> 4096 points per block
#define LEVELS  16
#define TBL     16384        // T, power of two -> hash mod is an AND
#define TMASK   (TBL - 1)

// ---- TDM path selection (probe-ordered; final fallback = async-to-LDS asm) ----
#if defined(__has_builtin)
#if __has_builtin(__builtin_amdgcn_tensor_load_to_lds_d2)
#define TDM_D2 1
#elif __has_builtin(__builtin_amdgcn_tensor_load_to_lds) && \
      __has_include(<hip/amd_detail/amd_gfx1250_TDM.h>)
#define TDM_6ARG 1
#elif __has_builtin(__builtin_amdgcn_tensor_load_to_lds)
#define TDM_5ARG 1
#endif
#endif

#if defined(TDM_D2) || defined(TDM_6ARG) || defined(TDM_5ARG)
#define USE_TDM 1
typedef unsigned int u32x4 __attribute__((ext_vector_type(4)));
typedef int          i32x8 __attribute__((ext_vector_type(8)));
typedef int          i32x4 __attribute__((ext_vector_type(4)));
#endif

// next_prime(2^17), next_prime(2^18) evaluated by the compiler (exact, no hand math)
constexpr bool is_prime_ce(unsigned k) {
  if (k < 2) return false;
  for (unsigned i = 2; i * i <= k; ++i)
    if (k % i == 0) return false;
  return true;
}
constexpr unsigned next_prime_ce(unsigned n) {
  while (!is_prime_ce(n)) ++n;
  return n;
}
constexpr unsigned P1 = next_prime_ce(1u << 17);
constexpr unsigned P2 = next_prime_ce(1u << 18);

struct ResArr { int r[LEVELS]; };   // per-level grid resolutions, passed by value

// One block: one level (blockIdx.y) x 4096 points (blockIdx.x).
// Stages the level's 128 KB latent table into LDS (TDM if available, else
// gfx1250 async copies), then all 8-corner gathers are ds_load_b64 from LDS.
__global__ __launch_bounds__(BLOCK)
void hashgrid_fwd(const float* __restrict__ pos,
                  const float* __restrict__ latents,
                  float* __restrict__ out,
                  ResArr res, int npts)
{
  extern __shared__ float2 smem[];            // TBL entries = 128 KB
  const int l   = blockIdx.y;
  const int tid = threadIdx.x;

  // ---- stage this level's 128 KB hash table into LDS ----
  const float2* table = reinterpret_cast<const float2*>(latents) + (size_t)l * TBL;
  // Flat LDS address carries the LDS byte offset in its low 32 bits.
  uint32_t lds_base = (uint32_t)(uintptr_t)(void*)smem;
  uint64_t gbase    = (uint64_t)(uintptr_t)(const void*)table;

#ifdef USE_TDM
  // Single Tensor-Data-Mover descriptor: 16384 x 8B contiguous tile -> LDS.
  if (tid < 32) {                    // wave 0 only (uniform branch per wave)
    const uint32_t lb  = __builtin_amdgcn_readfirstlane(lds_base);
    const uint32_t glo = __builtin_amdgcn_readfirstlane((uint32_t)gbase);
    const uint32_t ghi = __builtin_amdgcn_readfirstlane((uint32_t)(gbase >> 32));
    u32x4 g0;
    g0[0] = 1u;                                      // count=1, is_restore=0, gather=0
    g0[1] = lb;                                      // lds_addr[31:0]
    g0[2] = glo;                                     // global_addr[31:0]
    g0[3] = (ghi & 0x01FFFFFFu) | 0x80000000u;       // global_addr[56:32] | type=2
    i32x8 g1;
    g1[0] = 0x00030000;      // workgroup_mask=0 | data_size=3 (8B) | no barrier/iter/pad
    g1[1] = 0x40000000;      // tensor_dim0[15:0]=16384 in bits[31:16]
    g1[2] = 0x00010000;      // tensor_dim0[31:16]=0 | tensor_dim1[15:0]=1
    g1[3] = 0x40000000;      // tensor_dim1[31:16]=0 | tile_dim0=16384
    g1[4] = 0x00000001;      // tile_dim1=1 | tile_dim2=0 (unused)
    g1[5] = 16384;           // tensor_dim0_stride[31:0] (data_size units)
    g1[6] = 0;               // stride hi | tensor_dim1_stride lo
    g1[7] = 0;
#if defined(TDM_D2)
    __builtin_amdgcn_tensor_load_to_lds_d2(g0, g1, 0);
#elif defined(TDM_6ARG)
    i32x4 z4; z4[0] = 0; z4[1] = 0; z4[2] = 0; z4[3] = 0;
    i32x8 z8; z8[0] = 0; z8[1] = 0; z8[2] = 0; z8[3] = 0;
              z8[4] = 0; z8[5] = 0; z8[6] = 0; z8[7] = 0;
    __builtin_amdgcn_tensor_load_to_lds(g0, g1, z4, z4, z8, 0);
#else  // TDM_5ARG
    i32x4 z4; z4[0] = 0; z4[1] = 0; z4[2] = 0; z4[3] = 0;
    __builtin_amdgcn_tensor_load_to_lds(g0, g1, z4, z4, 0);
#endif
#if __has_builtin(__builtin_amdgcn_s_wait_tensorcnt)
    __builtin_amdgcn_s_wait_tensorcnt(0);
#else
    asm volatile("s_wait_tensorcnt 0x0" ::: "memory");
#endif
  }
#else
  constexpr int STAGE_ITERS = (TBL * (int)sizeof(float2)) / (BLOCK * 16);  // 32
  #pragma unroll
  for (int i = 0; i < STAGE_ITERS; ++i) {
    uint32_t off = (uint32_t)(i * BLOCK + tid) * 16u;
    uint32_t la  = lds_base + off;
    uint64_t ga  = gbase + off;
    asm volatile("global_load_async_to_lds_b128 %0, %1, off"
                 :: "v"(la), "v"(ga) : "memory");
  }
  asm volatile("s_wait_asynccnt 0" ::: "memory");
#endif
  __syncthreads();

  const float resf = (float)res.r[l];
  const int   base = blockIdx.x * (BLOCK * ITER);

  #pragma unroll 2
  for (int it = 0; it < ITER; ++it) {
    const int n = base + it * BLOCK + tid;
    if (n >= npts) break;

    const float px = pos[3 * n + 0];
    const float py = pos[3 * n + 1];
    const float pz = pos[3 * n + 2];

    const float sx = px * resf, sy = py * resf, sz = pz * resf;
    const float fx0 = floorf(sx), fy0 = floorf(sy), fz0 = floorf(sz);
    const int   ix = (int)fx0,   iy = (int)fy0,   iz = (int)fz0;
    const float fx = sx - fx0,   fy = sy - fy0,   fz = sz - fz0;

    // hash partials (prime for dim0 is 1)
    const uint32_t hy0 = (uint32_t)iy       * P1;
    const uint32_t hy1 = (uint32_t)(iy + 1) * P1;
    const uint32_t hz0 = (uint32_t)iz       * P2;
    const uint32_t hz1 = (uint32_t)(iz + 1) * P2;

    const float wx1 = fx, wx0 = 1.0f - fx;
    const float wy1 = fy, wy0 = 1.0f - fy;
    const float wz1 = fz, wz0 = 1.0f - fz;

    float ax = 0.0f, ay = 0.0f;
    #pragma unroll
    for (int c = 0; c < 8; ++c) {
      const int bx = (c >> 2) & 1, by = (c >> 1) & 1, bz = c & 1;
      const uint32_t h = (uint32_t)(ix + bx) ^ (by ? hy1 : hy0) ^ (bz ? hz1 : hz0);
      const float    w = (bx ? wx1 : wx0) * (by ? wy1 : wy0) * (bz ? wz1 : wz0);
      const float2   v = smem[h & TMASK];          // ds_load_b64 gather
      ax = fmaf(w, v.x, ax);
      ay = fmaf(w, v.y, ay);
    }

    float2 o; o.x = ax; o.y = ay;
    // out[n, l*2 .. l*2+1], 8-byte aligned
    *reinterpret_cast<float2*>(out + (size_t)n * (LEVELS * 2) + 2 * l) = o;
  }
}

extern "C" void kernel_launch(void* const* d_in, const int* in_sizes, int n_in,
                              void* d_out, int out_size, void* d_ws, size_t ws_size,
                              hipStream_t stream) {
  (void)n_in; (void)out_size; (void)d_ws; (void)ws_size;
  const float* pos     = (const float*)d_in[0];
  const float* latents = (const float*)d_in[1];
  float*       out     = (float*)d_out;
  const int    npts    = in_sizes[0] / 3;

  // Same double-precision math as the Python reference for per-level resolutions.
  ResArr res;
  const double b = std::exp((std::log(2048.0) - std::log(128.0)) / 15.0);
  for (int i = 0; i < LEVELS; ++i)
    res.r[i] = (int)(128.0 * std::pow(b, (double)i));

  const int    ppb   = BLOCK * ITER;
  const dim3   grid((npts + ppb - 1) / ppb, LEVELS, 1);
  const dim3   block(BLOCK, 1, 1);
  const size_t shmem = (size_t)TBL * sizeof(float2);   // 128 KB (<= 320 KB WGP LDS)

  // Opt in to >64 KB dynamic LDS (no-op if not required on this runtime).
  hipFuncSetAttribute(reinterpret_cast<const void*>(hashgrid_fwd),
                      hipFuncAttributeMaxDynamicSharedMemorySize, (int)shmem);

  hashgrid_fwd<<<grid, block, shmem, stream>>>(pos, latents, out, res, npts);
}